// DFTParallelRecon_diff_32804960207416
// MI455X (gfx1250) — compile-verified
//
#include <hip/hip_runtime.h>
#include <math.h>

// ---------------------------------------------------------------------------
// DFT-based fan-beam recon for MI455X (gfx1250, wave32, WMMA).
// Pipeline: real x complex GEMM (chirp-DFT) -> polar resample -> two complex
// GEMMs (chirp 2D iDFT, cropped).  GEMMs use V_WMMA_F32_16X16X4_F32, 2x2
// register tiles per wave, A row-major / B transposed in LDS so every
// fragment is a single aligned ds_load_b64 (no repack movs before wmma).
// ---------------------------------------------------------------------------

typedef __attribute__((ext_vector_type(2))) float v2f;
typedef __attribute__((ext_vector_type(8))) float v8f;

#define FN   2048   // padded detector FFT length
#define NDET 512
#define NV   512
#define FM   2048
#define MIMG 512
#define LO   768    // (FM - MIMG)/2

__device__ __forceinline__ v8f wmma4(v2f a, v2f b, v8f c) {
  // D = A(16x4 f32) * B(4x16 f32) + C(16x16 f32)
  return __builtin_amdgcn_wmma_f32_16x16x4_f32(
      false, a, false, b, (short)0, c, false, false);
}

// phase = exp(i * 2*pi * num/8192), scaled by amp.  num in int32 (exact).
__device__ __forceinline__ void phase8192(int num, float amp, float* cr, float* ci) {
  int r = num & 8191;
  float ang = (float)r * (6.28318530717958647692f / 8192.0f);
  float s, c;
  __sincosf(ang, &s, &c);
  *cr = amp * c;
  *ci = amp * s;
}

// ---------------------------------------------------------------------------
// Twiddle generators (all phases are exact rationals with denominator 8192).
// ---------------------------------------------------------------------------

// E1[m,n] = pre1[m]*post1[n]*exp(-2i pi m n / FN), m<512, n<2048.
__global__ void gen_e1(float* __restrict__ ere, float* __restrict__ eim) {
  int idx = blockIdx.x * 256 + threadIdx.x;   // 512*2048
  int m = idx >> 11;
  int n = idx & 2047;
  int num = 4094 * m + 1022 * n - 4 * m * n - 1046017;
  float cr, ci; phase8192(num, 1.0f, &cr, &ci);
  ere[idx] = cr; eim[idx] = ci;
}

// Fc'[v,x] = dk2^2 * post2[xg] * exp(+2i pi v xg / FM), xg = LO + x.  [2048x512]
__global__ void gen_fc(float* __restrict__ fre, float* __restrict__ fim) {
  int idx = blockIdx.x * 256 + threadIdx.x;   // 2048*512
  int v = idx >> 9;
  int x = idx & 511;
  int xg = LO + x;
  int num = 4 * v * xg - 4094 * xg + 4190209;
  float cr, ci; phase8192(num, 9.5367431640625e-7f, &cr, &ci);
  fre[idx] = cr; fim[idx] = ci;
}

// Fr'[y,u] = post2[yg] * exp(+2i pi u yg / FM), yg = LO + y.  [512 x 2048]
__global__ void gen_fr(float* __restrict__ fre, float* __restrict__ fim) {
  int idx = blockIdx.x * 256 + threadIdx.x;   // 512*2048
  int y = idx >> 11;
  int u = idx & 2047;
  int yg = LO + y;
  int num = 4 * u * yg - 4094 * yg + 4190209;
  float cr, ci; phase8192(num, 1.0f, &cr, &ci);
  fre[idx] = cr; fim[idx] = ci;
}

// ---------------------------------------------------------------------------
// Stage 1: Y[1024 x 2048] = sgm[1024 x 512] (real) * E1[512 x 2048] (complex)
// Block tile 64(M) x 128(N); wave tile 32x32 (2x2 WMMA tiles).
// ---------------------------------------------------------------------------
__global__ __launch_bounds__(256) void gemm_rc(
    const float* __restrict__ A,
    const float* __restrict__ Bre, const float* __restrict__ Bim,
    float* __restrict__ Cre, float* __restrict__ Cim) {
  const int Mtile = blockIdx.y * 64;
  const int Ntile = blockIdx.x * 128;
  const int tid  = threadIdx.x;
  const int lane = tid & 31;
  const int wave = tid >> 5;
  const int wmB = (wave >> 2) * 32;   // 0 or 32
  const int wnB = (wave & 3) * 32;    // 0,32,64,96
  const int lh = lane >> 4;
  const int ll = lane & 15;

  __shared__ float As[64][66];         // [m][k], pad 2: 8B-aligned v2f reads
  __shared__ float Bst[2][128][66];    // B transposed: [plane][n][k]

  v8f accR[2][2] = {}; v8f accI[2][2] = {};

  for (int ks = 0; ks < 512; ks += 64) {
    __syncthreads();
    for (int t = tid; t < 64 * 64; t += 256) {
      int m = t >> 6, c = t & 63;
      As[m][c] = A[(size_t)(Mtile + m) * 512 + ks + c];
    }
    for (int t = tid; t < 64 * 128; t += 256) {
      int k = t >> 7, c = t & 127;
      size_t g = (size_t)(ks + k) * 2048 + Ntile + c;
      Bst[0][c][k] = Bre[g];
      Bst[1][c][k] = Bim[g];
    }
    __syncthreads();
    if (ks + 64 < 512) {   // prefetch next K-slab (global_prefetch_b8)
      __builtin_prefetch(&A[(size_t)(Mtile + (tid >> 6)) * 512 + ks + 64 + (tid & 63)], 0, 0);
      __builtin_prefetch(&Bre[(size_t)(ks + 64 + (tid >> 7)) * 2048 + Ntile + (tid & 127)], 0, 0);
      __builtin_prefetch(&Bim[(size_t)(ks + 64 + (tid >> 7)) * 2048 + Ntile + (tid & 127)], 0, 0);
    }
#pragma unroll
    for (int kk = 0; kk < 64; kk += 4) {
      int kr = kk + 2 * lh;
      v2f a[2], br[2], bi[2];
#pragma unroll
      for (int t = 0; t < 2; ++t) {
        a[t]  = *(const v2f*)&As[wmB + t * 16 + ll][kr];
        br[t] = *(const v2f*)&Bst[0][wnB + t * 16 + ll][kr];
        bi[t] = *(const v2f*)&Bst[1][wnB + t * 16 + ll][kr];
      }
#pragma unroll
      for (int mi = 0; mi < 2; ++mi)
#pragma unroll
        for (int ni = 0; ni < 2; ++ni) {
          accR[mi][ni] = wmma4(a[mi], br[ni], accR[mi][ni]);
          accI[mi][ni] = wmma4(a[mi], bi[ni], accI[mi][ni]);
        }
    }
  }
#pragma unroll
  for (int mi = 0; mi < 2; ++mi)
#pragma unroll
    for (int ni = 0; ni < 2; ++ni)
#pragma unroll
      for (int r = 0; r < 8; ++r) {
        int m = Mtile + wmB + mi * 16 + r + 8 * lh;
        int n = Ntile + wnB + ni * 16 + ll;
        Cre[(size_t)m * 2048 + n] = accR[mi][ni][r];
        Cim[(size_t)m * 2048 + n] = accI[mi][ni][r];
      }
}

// ---------------------------------------------------------------------------
// Stage 2: polar -> Cartesian bilinear resample + pre-chirp of 2D iDFT.
// ---------------------------------------------------------------------------
__global__ void polar_resample(const float* __restrict__ yre,
                               const float* __restrict__ yim,
                               float* __restrict__ gre,
                               float* __restrict__ gim) {
  int idx = blockIdx.x * 256 + threadIdx.x;   // FM*FM
  int i = idx >> 11;   // row -> kx
  int j = idx & 2047;  // col -> ky
  float kx = ((float)i - 1023.5f) * (1.0f / 1024.0f);
  float ky = ((float)j - 1023.5f) * (1.0f / 1024.0f);
  float mk = sqrtf(kx * kx + ky * ky);
  float th = atan2f(ky, kx);
  float ix = (2.0f * mk + 1.0f) * (0.5f * 2047.0f);
  float iy = (th * (1.0f / 6.28318530717958647692f) + 1.0f) * (0.5f * 1023.0f);

  float x0f = floorf(ix), y0f = floorf(iy);
  int   x0 = (int)x0f,   y0 = (int)y0f;
  float wx1 = ix - x0f,  wy1 = iy - y0f;

  float sre0 = 0.f, sim0 = 0.f, sre1 = 0.f, sim1 = 0.f;
#pragma unroll
  for (int dy = 0; dy < 2; ++dy) {
    int yi = y0 + dy;
    float wy = dy ? wy1 : (1.0f - wy1);
    bool vy = (yi >= 0) && (yi < 2 * NV);
    int yc = min(max(yi, 0), 2 * NV - 1) & (NV - 1);   // angular wrap
#pragma unroll
    for (int dx = 0; dx < 2; ++dx) {
      int xi = x0 + dx;
      float wx = dx ? wx1 : (1.0f - wx1);
      bool vx = (xi >= 0) && (xi < FN);
      int xc = min(max(xi, 0), FN - 1);
      float w = (vx && vy) ? (wx * wy) : 0.0f;
      size_t o0 = ((size_t)yc << 11) + xc;             // batch 0
      size_t o1 = (((size_t)(NV + yc)) << 11) + xc;    // batch 1
      sre0 += w * yre[o0]; sim0 += w * yim[o0];
      sre1 += w * yre[o1]; sim1 += w * yim[o1];
    }
  }
  float c, s; phase8192(-4094 * (i + j), 1.0f, &c, &s);
  size_t o = (size_t)idx;
  gre[o] = sre0 * c - sim0 * s;
  gim[o] = sre0 * s + sim0 * c;
  gre[o + (size_t)FM * FM] = sre1 * c - sim1 * s;
  gim[o + (size_t)FM * FM] = sre1 * s + sim1 * c;
}

// ---------------------------------------------------------------------------
// Stage 3a: T[4096 x 512] = gm2[4096 x 2048] * Fc'[2048 x 512]  (complex)
// ---------------------------------------------------------------------------
__global__ __launch_bounds__(256) void gemm_cc(
    const float* __restrict__ Are, const float* __restrict__ Aim,
    const float* __restrict__ Bre, const float* __restrict__ Bim,
    float* __restrict__ Cre, float* __restrict__ Cim) {
  const int Mtile = blockIdx.y * 64;
  const int Ntile = blockIdx.x * 128;
  const int tid  = threadIdx.x;
  const int lane = tid & 31;
  const int wave = tid >> 5;
  const int wmB = (wave >> 2) * 32;
  const int wnB = (wave & 3) * 32;
  const int lh = lane >> 4;
  const int ll = lane & 15;

  __shared__ float As[2][64][66];
  __shared__ float Bst[2][128][66];    // B transposed: [plane][n][k]

  v8f RR[2][2] = {}, II[2][2] = {}, RI[2][2] = {}, IR[2][2] = {};

  for (int ks = 0; ks < 2048; ks += 64) {
    __syncthreads();
    for (int t = tid; t < 64 * 64; t += 256) {
      int m = t >> 6, c = t & 63;
      size_t g = (size_t)(Mtile + m) * 2048 + ks + c;
      As[0][m][c] = Are[g];
      As[1][m][c] = Aim[g];
    }
    for (int t = tid; t < 64 * 128; t += 256) {
      int k = t >> 7, c = t & 127;
      size_t g = (size_t)(ks + k) * 512 + Ntile + c;
      Bst[0][c][k] = Bre[g];
      Bst[1][c][k] = Bim[g];
    }
    __syncthreads();
    if (ks + 64 < 2048) {
      __builtin_prefetch(&Are[(size_t)(Mtile + (tid >> 6)) * 2048 + ks + 64 + (tid & 63)], 0, 0);
      __builtin_prefetch(&Aim[(size_t)(Mtile + (tid >> 6)) * 2048 + ks + 64 + (tid & 63)], 0, 0);
      __builtin_prefetch(&Bre[(size_t)(ks + 64 + (tid >> 7)) * 512 + Ntile + (tid & 127)], 0, 0);
      __builtin_prefetch(&Bim[(size_t)(ks + 64 + (tid >> 7)) * 512 + Ntile + (tid & 127)], 0, 0);
    }
#pragma unroll
    for (int kk = 0; kk < 64; kk += 4) {
      int kr = kk + 2 * lh;
      v2f ar[2], ai[2], br[2], bi[2];
#pragma unroll
      for (int t = 0; t < 2; ++t) {
        ar[t] = *(const v2f*)&As[0][wmB + t * 16 + ll][kr];
        ai[t] = *(const v2f*)&As[1][wmB + t * 16 + ll][kr];
        br[t] = *(const v2f*)&Bst[0][wnB + t * 16 + ll][kr];
        bi[t] = *(const v2f*)&Bst[1][wnB + t * 16 + ll][kr];
      }
#pragma unroll
      for (int mi = 0; mi < 2; ++mi)
#pragma unroll
        for (int ni = 0; ni < 2; ++ni) {
          RR[mi][ni] = wmma4(ar[mi], br[ni], RR[mi][ni]);
          II[mi][ni] = wmma4(ai[mi], bi[ni], II[mi][ni]);
          RI[mi][ni] = wmma4(ar[mi], bi[ni], RI[mi][ni]);
          IR[mi][ni] = wmma4(ai[mi], br[ni], IR[mi][ni]);
        }
    }
  }
#pragma unroll
  for (int mi = 0; mi < 2; ++mi)
#pragma unroll
    for (int ni = 0; ni < 2; ++ni)
#pragma unroll
      for (int r = 0; r < 8; ++r) {
        int m = Mtile + wmB + mi * 16 + r + 8 * lh;
        int n = Ntile + wnB + ni * 16 + ll;
        Cre[(size_t)m * 512 + n] = RR[mi][ni][r] - II[mi][ni][r];
        Cim[(size_t)m * 512 + n] = RI[mi][ni][r] + IR[mi][ni][r];
      }
}

// ---------------------------------------------------------------------------
// Stage 3b: out[b][512 x 512] = Re( Fr'[512 x 2048] * T_b[2048 x 512] )
// ---------------------------------------------------------------------------
__global__ __launch_bounds__(256) void gemm_cr(
    const float* __restrict__ Are, const float* __restrict__ Aim,
    const float* __restrict__ Bre, const float* __restrict__ Bim,
    float* __restrict__ Out) {
  const int b = blockIdx.z;
  const size_t boffB = (size_t)b * 2048 * 512;
  const size_t boffO = (size_t)b * 512 * 512;
  const int Mtile = blockIdx.y * 64;
  const int Ntile = blockIdx.x * 128;
  const int tid  = threadIdx.x;
  const int lane = tid & 31;
  const int wave = tid >> 5;
  const int wmB = (wave >> 2) * 32;
  const int wnB = (wave & 3) * 32;
  const int lh = lane >> 4;
  const int ll = lane & 15;

  __shared__ float As[2][64][66];
  __shared__ float Bst[2][128][66];

  v8f P[2][2] = {}, Q[2][2] = {};

  for (int ks = 0; ks < 2048; ks += 64) {
    __syncthreads();
    for (int t = tid; t < 64 * 64; t += 256) {
      int m = t >> 6, c = t & 63;
      size_t g = (size_t)(Mtile + m) * 2048 + ks + c;
      As[0][m][c] = Are[g];
      As[1][m][c] = Aim[g];
    }
    for (int t = tid; t < 64 * 128; t += 256) {
      int k = t >> 7, c = t & 127;
      size_t g = boffB + (size_t)(ks + k) * 512 + Ntile + c;
      Bst[0][c][k] = Bre[g];
      Bst[1][c][k] = Bim[g];
    }
    __syncthreads();
    if (ks + 64 < 2048) {
      __builtin_prefetch(&Are[(size_t)(Mtile + (tid >> 6)) * 2048 + ks + 64 + (tid & 63)], 0, 0);
      __builtin_prefetch(&Aim[(size_t)(Mtile + (tid >> 6)) * 2048 + ks + 64 + (tid & 63)], 0, 0);
      __builtin_prefetch(&Bre[boffB + (size_t)(ks + 64 + (tid >> 7)) * 512 + Ntile + (tid & 127)], 0, 0);
      __builtin_prefetch(&Bim[boffB + (size_t)(ks + 64 + (tid >> 7)) * 512 + Ntile + (tid & 127)], 0, 0);
    }
#pragma unroll
    for (int kk = 0; kk < 64; kk += 4) {
      int kr = kk + 2 * lh;
      v2f ar[2], ai[2], br[2], bi[2];
#pragma unroll
      for (int t = 0; t < 2; ++t) {
        ar[t] = *(const v2f*)&As[0][wmB + t * 16 + ll][kr];
        ai[t] = *(const v2f*)&As[1][wmB + t * 16 + ll][kr];
        br[t] = *(const v2f*)&Bst[0][wnB + t * 16 + ll][kr];
        bi[t] = *(const v2f*)&Bst[1][wnB + t * 16 + ll][kr];
      }
#pragma unroll
      for (int mi = 0; mi < 2; ++mi)
#pragma unroll
        for (int ni = 0; ni < 2; ++ni) {
          P[mi][ni] = wmma4(ar[mi], br[ni], P[mi][ni]);
          Q[mi][ni] = wmma4(ai[mi], bi[ni], Q[mi][ni]);
        }
    }
  }
#pragma unroll
  for (int mi = 0; mi < 2; ++mi)
#pragma unroll
    for (int ni = 0; ni < 2; ++ni)
#pragma unroll
      for (int r = 0; r < 8; ++r) {
        int m = Mtile + wmB + mi * 16 + r + 8 * lh;
        int n = Ntile + wnB + ni * 16 + ll;
        Out[boffO + (size_t)m * 512 + n] = P[mi][ni][r] - Q[mi][ni][r];
      }
}

// ---------------------------------------------------------------------------
extern "C" void kernel_launch(void* const* d_in, const int* in_sizes, int n_in,
                              void* d_out, int out_size, void* d_ws, size_t ws_size,
                              hipStream_t stream) {
  (void)in_sizes; (void)n_in; (void)out_size; (void)ws_size;
  const float* sgm = (const float*)d_in[0];   // [2,1,512,512] -> [1024 x 512]
  float* out = (float*)d_out;                 // [2,1,512,512]
  float* ws  = (float*)d_ws;

  // workspace layout (floats); total ~31.5M floats = ~120 MB
  const size_t oE1re = 0;
  const size_t oE1im = oE1re + (size_t)512 * 2048;
  const size_t oYre  = oE1im + (size_t)512 * 2048;
  const size_t oYim  = oYre  + (size_t)1024 * 2048;
  const size_t oFcre = oYim  + (size_t)1024 * 2048;
  const size_t oFcim = oFcre + (size_t)2048 * 512;
  const size_t oFrre = oFcim + (size_t)2048 * 512;
  const size_t oFrim = oFrre + (size_t)512 * 2048;
  const size_t oGre  = oFrim + (size_t)512 * 2048;
  const size_t oGim  = oGre  + (size_t)2 * 2048 * 2048;
  const size_t oTre  = oGim  + (size_t)2 * 2048 * 2048;
  const size_t oTim  = oTre  + (size_t)2 * 2048 * 512;

  gen_e1<<<(512 * 2048) / 256, 256, 0, stream>>>(ws + oE1re, ws + oE1im);
  gen_fc<<<(2048 * 512) / 256, 256, 0, stream>>>(ws + oFcre, ws + oFcim);
  gen_fr<<<(512 * 2048) / 256, 256, 0, stream>>>(ws + oFrre, ws + oFrim);

  gemm_rc<<<dim3(2048 / 128, 1024 / 64), 256, 0, stream>>>(
      sgm, ws + oE1re, ws + oE1im, ws + oYre, ws + oYim);

  polar_resample<<<(2048 * 2048) / 256, 256, 0, stream>>>(
      ws + oYre, ws + oYim, ws + oGre, ws + oGim);

  gemm_cc<<<dim3(512 / 128, 4096 / 64), 256, 0, stream>>>(
      ws + oGre, ws + oGim, ws + oFcre, ws + oFcim, ws + oTre, ws + oTim);

  gemm_cr<<<dim3(512 / 128, 512 / 64, 2), 256, 0, stream>>>(
      ws + oFrre, ws + oFrim, ws + oTre, ws + oTim, out);
}